// NeuralSparseCode_31568009626113
// MI455X (gfx1250) — compile-verified
//
#include <hip/hip_runtime.h>
#include <hip/hip_bf16.h>

#define Bb 4
#define Cc 128
#define Tn 4096
#define NA 4096
#define KS 128
#define NB_HIST 2048
#define CAP 8192

typedef float v2f __attribute__((ext_vector_type(2)));
typedef float v8f __attribute__((ext_vector_type(8)));

__device__ __forceinline__ v8f wmma_f32_k4(v2f a, v2f b, v8f c) {
  // D = A(16x4 f32) * B(4x16 f32) + C(16x16 f32)
  return __builtin_amdgcn_wmma_f32_16x16x4_f32(false, a, false, b, (short)0, c,
                                               false, false);
}

// Async global->LDS copy (CDNA5 GLOBAL_LOAD_ASYNC_TO_LDS_B32, ASYNCcnt).
// Generic __shared__ pointers carry the wave-relative LDS byte offset in the
// low 32 bits (flat-addressing LDS aperture rules), which is what VDST wants.
__device__ __forceinline__ void async_copy_b32(void* lds_dst,
                                               const float* gsrc) {
  const unsigned loff = (unsigned)(unsigned long long)lds_dst;
  asm volatile("global_load_async_to_lds_b32 %0, %1, off" ::"v"(loff),
               "v"(gsrc)
               : "memory");
}
__device__ __forceinline__ void wait_async0() {
  asm volatile("s_wait_asynccnt 0" ::: "memory");
}

// ---------------------------------------------------------------------------
// Dilated conv3 (pad=dil) + LeakyReLU(0.2) + per-channel sum/sumsq for BN.
// Optional per-channel scale/shift folded into the input load (BN of the
// previous layer). One block = 8 waves; wave w computes cout tile [16w,16w+16)
// for a 16-wide time tile; input tile staged in LDS.
// ---------------------------------------------------------------------------
template <int DIL>
__global__ __launch_bounds__(256) void conv_k3_lrelu_stats(
    const float* __restrict__ x, const float* __restrict__ w,
    const float* __restrict__ bias, const float* __restrict__ scale,
    const float* __restrict__ shift, float* __restrict__ y,
    float* __restrict__ gsum, float* __restrict__ gsq) {
  constexpr int SPAN = 16 + 2 * DIL;
  __shared__ float lx[Cc * SPAN];
  __shared__ float lsum[Cc], lsq[Cc];
  const int tid = threadIdx.x;
  const int tile = blockIdx.x;  // B * T/16 tiles
  const int b = tile >> 8;
  const int t0 = (tile & 255) << 4;

  // L2 prefetch of the (L2-resident, all-blocks-shared) weight tensor.
  __builtin_prefetch(w + (size_t)tid * 192, 0, 1);

  if (tid < Cc) { lsum[tid] = 0.f; lsq[tid] = 0.f; }
  for (int i = tid; i < Cc * SPAN; i += 256) {
    const int c = i / SPAN, col = i % SPAN;
    const int g = t0 - DIL + col;
    float v = 0.f;
    if (g >= 0 && g < Tn) v = x[(b * Cc + c) * Tn + g];
    if (scale) v = v * scale[c] + shift[c];
    lx[i] = v;
  }
  __syncthreads();

  const int wave = tid >> 5, lane = tid & 31;
  const int m = lane & 15;          // A-matrix row (M)
  const int kk = (lane >> 4) << 1;  // K pair: lanes 0-15 -> K{0,1}, hi -> K{2,3}
  const int n = lane & 15;          // B/C/D column (N)
  const int cout0 = wave << 4;

  v8f acc = {};
#pragma unroll
  for (int tap = 0; tap < 3; ++tap) {
    const int base = tap * DIL + n;
#pragma unroll 4
    for (int c0 = 0; c0 < Cc; c0 += 4) {
      v2f a, bv;
      a.x = w[(cout0 + m) * (Cc * 3) + (c0 + kk) * 3 + tap];
      a.y = w[(cout0 + m) * (Cc * 3) + (c0 + kk + 1) * 3 + tap];
      bv.x = lx[(c0 + kk) * SPAN + base];
      bv.y = lx[(c0 + kk + 1) * SPAN + base];
      acc = wmma_f32_k4(a, bv, acc);
    }
  }

#pragma unroll
  for (int r = 0; r < 8; ++r) {
    const int cout = cout0 + r + ((lane >> 4) << 3);
    float v = acc[r] + bias[cout];
    v = v > 0.f ? v : 0.2f * v;  // LeakyReLU(0.2)
    y[(b * Cc + cout) * Tn + t0 + n] = v;
    atomicAdd(&lsum[cout], v);
    atomicAdd(&lsq[cout], v * v);
  }
  __syncthreads();
  if (tid < Cc) {
    atomicAdd(&gsum[tid], lsum[tid]);
    atomicAdd(&gsq[tid], lsq[tid]);
  }
}

__global__ void bn_finalize(const float* __restrict__ gsum,
                            const float* __restrict__ gsq,
                            const float* __restrict__ gamma,
                            const float* __restrict__ beta,
                            float* __restrict__ scale,
                            float* __restrict__ shift) {
  const int c = threadIdx.x;
  const float n = (float)(Bb * Tn);
  const float mean = gsum[c] / n;
  const float var = gsq[c] / n - mean * mean;
  const float s = gamma[c] * rsqrtf(var + 1e-5f);
  scale[c] = s;
  shift[c] = beta[c] - mean * s;
}

// out = skip + bn(raw)   (encoder residual)
__global__ __launch_bounds__(256) void bn_apply_add(
    const float* __restrict__ raw, const float* __restrict__ scale,
    const float* __restrict__ shift, const float* __restrict__ skip,
    float* __restrict__ out) {
  const int i = blockIdx.x * 256 + threadIdx.x;
  if (i < Bb * Cc * Tn) {
    const int c = (i / Tn) & (Cc - 1);
    out[i] = skip[i] + raw[i] * scale[c] + shift[c];
  }
}

__global__ void zero_u32(unsigned* __restrict__ p, int n) {
  const int i = blockIdx.x * 256 + threadIdx.x;
  if (i < n) p[i] = 0u;
}

__device__ __forceinline__ unsigned order_key(float v) {
  const unsigned u = __float_as_uint(v);
  return (u & 0x80000000u) ? ~u : (u | 0x80000000u);
}

// ---------------------------------------------------------------------------
// Up-conv (1x1, 128 -> 4096 atoms) fused with order-key histogram. The 256 MB
// code tensor is never materialized. Block: 8 waves x 16 atoms, one 16-wide
// time tile. Input tile staged via async global->LDS DMA.
// ---------------------------------------------------------------------------
__global__ __launch_bounds__(256) void up_hist(
    const float* __restrict__ x, const float* __restrict__ w,
    const float* __restrict__ bias, unsigned* __restrict__ hist) {
  __shared__ float lx[Cc * 16];
  __shared__ unsigned lh[NB_HIST];
  const int tid = threadIdx.x;
  const int tile = blockIdx.x;  // 32 atom-groups * 1024 bt-tiles
  const int ag = tile >> 10;
  const int bt = tile & 1023;
  const int b = bt >> 8;
  const int t0 = (bt & 255) << 4;
  for (int i = tid; i < Cc * 16; i += 256)  // 2048 = 8 * 256, no divergence
    async_copy_b32(&lx[i], &x[(b * Cc + (i >> 4)) * Tn + t0 + (i & 15)]);
  for (int i = tid; i < NB_HIST; i += 256) lh[i] = 0u;
  wait_async0();
  __syncthreads();

  const int wave = tid >> 5, lane = tid & 31;
  const int m = lane & 15, kk = (lane >> 4) << 1, n = lane & 15;
  const int a0 = (ag * 8 + wave) << 4;
  v8f acc = {};
#pragma unroll 4
  for (int c0 = 0; c0 < Cc; c0 += 4) {
    v2f a, bv;
    a.x = w[(a0 + m) * Cc + c0 + kk];
    a.y = w[(a0 + m) * Cc + c0 + kk + 1];
    bv.x = lx[(c0 + kk) * 16 + n];
    bv.y = lx[(c0 + kk + 1) * 16 + n];
    acc = wmma_f32_k4(a, bv, acc);
  }
#pragma unroll
  for (int r = 0; r < 8; ++r) {
    const int atom = a0 + r + ((lane >> 4) << 3);
    const unsigned key = order_key(acc[r] + bias[atom]);
    atomicAdd(&lh[key >> 21], 1u);
  }
  __syncthreads();
  for (int i = tid; i < NB_HIST; i += 256)
    if (lh[i]) atomicAdd(&hist[b * NB_HIST + i], lh[i]);
}

__global__ void topk_threshold(const unsigned* __restrict__ hist,
                               unsigned* __restrict__ thresh,
                               unsigned* __restrict__ ccnt) {
  const int b = blockIdx.x;
  if (threadIdx.x == 0) {
    unsigned s = 0;
    int tb = 0;
    for (int i = NB_HIST - 1; i >= 0; --i) {
      s += hist[b * NB_HIST + i];
      if (s >= (unsigned)KS) { tb = i; break; }
    }
    thresh[b] = ((unsigned)tb) << 21;
    ccnt[b] = 0u;
  }
}

// Recompute up-conv, collect only candidates (key >= bucket threshold).
__global__ __launch_bounds__(256) void up_collect(
    const float* __restrict__ x, const float* __restrict__ w,
    const float* __restrict__ bias, const unsigned* __restrict__ thresh,
    unsigned* __restrict__ ccnt, unsigned* __restrict__ ckey,
    unsigned* __restrict__ cpos) {
  __shared__ float lx[Cc * 16];
  const int tid = threadIdx.x;
  const int tile = blockIdx.x;
  const int ag = tile >> 10;
  const int bt = tile & 1023;
  const int b = bt >> 8;
  const int t0 = (bt & 255) << 4;
  for (int i = tid; i < Cc * 16; i += 256)
    async_copy_b32(&lx[i], &x[(b * Cc + (i >> 4)) * Tn + t0 + (i & 15)]);
  wait_async0();
  __syncthreads();

  const int wave = tid >> 5, lane = tid & 31;
  const int m = lane & 15, kk = (lane >> 4) << 1, n = lane & 15;
  const int a0 = (ag * 8 + wave) << 4;
  v8f acc = {};
#pragma unroll 4
  for (int c0 = 0; c0 < Cc; c0 += 4) {
    v2f a, bv;
    a.x = w[(a0 + m) * Cc + c0 + kk];
    a.y = w[(a0 + m) * Cc + c0 + kk + 1];
    bv.x = lx[(c0 + kk) * 16 + n];
    bv.y = lx[(c0 + kk + 1) * 16 + n];
    acc = wmma_f32_k4(a, bv, acc);
  }
  const unsigned th = thresh[b];
#pragma unroll
  for (int r = 0; r < 8; ++r) {
    const int atom = a0 + r + ((lane >> 4) << 3);
    const unsigned key = order_key(acc[r] + bias[atom]);
    if (key >= th) {
      const unsigned p = atomicAdd(&ccnt[b], 1u);
      if (p < CAP) {
        ckey[b * CAP + p] = key;
        cpos[b * CAP + p] = ((unsigned)atom << 12) | (unsigned)(t0 + n);
      }
    }
  }
}

// Exact rank-select among candidates; emit the top-128 (atom,t,val) list.
__global__ __launch_bounds__(256) void topk_select(
    const unsigned* __restrict__ ckey, const unsigned* __restrict__ cpos,
    const unsigned* __restrict__ ccnt, unsigned* __restrict__ spos,
    float* __restrict__ sval) {
  const int b = blockIdx.x;
  const int nc = (int)(ccnt[b] > CAP ? CAP : ccnt[b]);
  for (int i = threadIdx.x; i < nc; i += 256) {
    const unsigned ki = ckey[b * CAP + i];
    int rank = 0;
    for (int j = 0; j < nc; ++j) {
      const unsigned kj = ckey[b * CAP + j];
      if (kj > ki || (kj == ki && j < i)) ++rank;
    }
    if (rank < KS) {
      spos[b * KS + rank] = cpos[b * CAP + i];
      const unsigned u = (ki & 0x80000000u) ? (ki & 0x7fffffffu) : ~ki;
      sval[b * KS + rank] = __uint_as_float(u);
    }
  }
}

// Sparse map conv: out = map_b broadcast, then scatter 128 atoms per batch.
__global__ __launch_bounds__(256) void map_init(float* __restrict__ out,
                                                const float* __restrict__ mb) {
  const int i = blockIdx.x * 256 + threadIdx.x;
  if (i < Bb * Cc * Tn) out[i] = mb[(i / Tn) & (Cc - 1)];
}

__global__ void map_scatter(const unsigned* __restrict__ spos,
                            const float* __restrict__ sval,
                            const float* __restrict__ mw,
                            float* __restrict__ out) {
  const int b = blockIdx.x;
  const int c = threadIdx.x;  // 128 threads
  for (int e = 0; e < KS; ++e) {
    const unsigned p = spos[b * KS + e];
    const float v = sval[b * KS + e];
    const int atom = (int)(p >> 12);
    const int t = (int)(p & 4095u);
    atomicAdd(&out[(b * Cc + c) * Tn + t], mw[c * NA + atom] * v);
  }
}

// Final 1x1 conv with BN of decoder-L4 folded into the load, + residual skip.
__global__ __launch_bounds__(256) void conv1x1_bn_skip(
    const float* __restrict__ raw, const float* __restrict__ w,
    const float* __restrict__ bias, const float* __restrict__ scale,
    const float* __restrict__ shift, const float* __restrict__ skip,
    float* __restrict__ out) {
  __shared__ float lx[Cc * 16];
  const int tid = threadIdx.x;
  const int tile = blockIdx.x;  // B*T/16
  const int b = tile >> 8;
  const int t0 = (tile & 255) << 4;
  for (int i = tid; i < Cc * 16; i += 256) {
    const int c = i >> 4;
    lx[i] = raw[(b * Cc + c) * Tn + t0 + (i & 15)] * scale[c] + shift[c];
  }
  __syncthreads();

  const int wave = tid >> 5, lane = tid & 31;
  const int m = lane & 15, kk = (lane >> 4) << 1, n = lane & 15;
  const int cout0 = wave << 4;
  v8f acc = {};
#pragma unroll 4
  for (int c0 = 0; c0 < Cc; c0 += 4) {
    v2f a, bv;
    a.x = w[(cout0 + m) * Cc + c0 + kk];
    a.y = w[(cout0 + m) * Cc + c0 + kk + 1];
    bv.x = lx[(c0 + kk) * 16 + n];
    bv.y = lx[(c0 + kk + 1) * 16 + n];
    acc = wmma_f32_k4(a, bv, acc);
  }
#pragma unroll
  for (int r = 0; r < 8; ++r) {
    const int cout = cout0 + r + ((lane >> 4) << 3);
    const int idx = (b * Cc + cout) * Tn + t0 + n;
    out[idx] = acc[r] + bias[cout] + skip[idx];
  }
}

// ---------------------------------------------------------------------------

extern "C" void kernel_launch(void* const* d_in, const int* in_sizes, int n_in,
                              void* d_out, int out_size, void* d_ws,
                              size_t ws_size, hipStream_t stream) {
  (void)in_sizes; (void)n_in; (void)out_size; (void)ws_size;
  const float* X = (const float*)d_in[0];
  // enc_params: 4 layers x (w,b,gamma,beta) at d_in[1..16]
  // up_w=17 up_b=18 map_w=19 map_b=20, dec_params at 21..36, out_w=37 out_b=38
  const float* UPW = (const float*)d_in[17];
  const float* UPB = (const float*)d_in[18];
  const float* MW = (const float*)d_in[19];
  const float* MB = (const float*)d_in[20];
  const float* OW = (const float*)d_in[37];
  const float* OB = (const float*)d_in[38];

  const int NACT = Bb * Cc * Tn;  // 2,097,152
  float* buf0 = (float*)d_ws;
  float* buf1 = buf0 + NACT;
  float* buf2 = buf1 + NACT;
  float* s_sum = buf2 + NACT;
  float* s_sq = s_sum + Cc;
  float* s_scale = s_sq + Cc;
  float* s_shift = s_scale + Cc;
  unsigned* hist = (unsigned*)(s_shift + Cc);  // 4*2048
  unsigned* thresh = hist + Bb * NB_HIST;      // 4
  unsigned* ccnt = thresh + Bb;                // 4
  unsigned* ckey = ccnt + Bb;                  // 4*CAP
  unsigned* cpos = ckey + Bb * CAP;            // 4*CAP
  unsigned* spos = cpos + Bb * CAP;            // 4*128
  float* sval = (float*)(spos + Bb * KS);      // 4*128

  const int convGrid = Bb * (Tn / 16);           // 1024
  const int upGrid = (NA / 128) * convGrid;      // 32768
  const int ewGrid = (NACT + 255) / 256;         // 8192

  auto run_layer = [&](auto kfn, const float* in, int li_base, int layer,
                       const float* sc, const float* sh, float* out) {
    zero_u32<<<1, 256, 0, stream>>>((unsigned*)s_sum, 2 * Cc);
    kfn<<<convGrid, 256, 0, stream>>>(
        in, (const float*)d_in[li_base + 4 * layer + 0],
        (const float*)d_in[li_base + 4 * layer + 1], sc, sh, out, s_sum, s_sq);
    bn_finalize<<<1, Cc, 0, stream>>>(
        s_sum, s_sq, (const float*)d_in[li_base + 4 * layer + 2],
        (const float*)d_in[li_base + 4 * layer + 3], s_scale, s_shift);
  };

  // ---- encoder block (dilations 1,3,9,27), BN folded into next load ----
  run_layer(conv_k3_lrelu_stats<1>, X, 1, 0, (const float*)nullptr,
            (const float*)nullptr, buf0);
  run_layer(conv_k3_lrelu_stats<3>, buf0, 1, 1, s_scale, s_shift, buf1);
  run_layer(conv_k3_lrelu_stats<9>, buf1, 1, 2, s_scale, s_shift, buf0);
  run_layer(conv_k3_lrelu_stats<27>, buf0, 1, 3, s_scale, s_shift, buf1);
  bn_apply_add<<<ewGrid, 256, 0, stream>>>(buf1, s_scale, s_shift, X, buf2);

  // ---- up-conv + fused top-128 (never materialize 256 MB codes) ----
  zero_u32<<<(Bb * NB_HIST + 255) / 256, 256, 0, stream>>>(hist, Bb * NB_HIST);
  zero_u32<<<(2 * Bb * KS + 255) / 256, 256, 0, stream>>>(spos, 2 * Bb * KS);
  up_hist<<<upGrid, 256, 0, stream>>>(buf2, UPW, UPB, hist);
  topk_threshold<<<Bb, 32, 0, stream>>>(hist, thresh, ccnt);
  up_collect<<<upGrid, 256, 0, stream>>>(buf2, UPW, UPB, thresh, ccnt, ckey,
                                         cpos);
  topk_select<<<Bb, 256, 0, stream>>>(ckey, cpos, ccnt, spos, sval);

  // ---- sparse map conv (128 nonzeros per batch) ----
  map_init<<<ewGrid, 256, 0, stream>>>(buf0, MB);
  map_scatter<<<Bb, Cc, 0, stream>>>(spos, sval, MW, buf0);

  // ---- decoder block ----
  run_layer(conv_k3_lrelu_stats<1>, buf0, 21, 0, (const float*)nullptr,
            (const float*)nullptr, buf1);
  run_layer(conv_k3_lrelu_stats<3>, buf1, 21, 1, s_scale, s_shift, buf2);
  run_layer(conv_k3_lrelu_stats<9>, buf2, 21, 2, s_scale, s_shift, buf1);
  run_layer(conv_k3_lrelu_stats<27>, buf1, 21, 3, s_scale, s_shift, buf2);

  // ---- final 1x1 conv (BN folded) + residual ----
  conv1x1_bn_skip<<<convGrid, 256, 0, stream>>>(buf2, OW, OB, s_scale, s_shift,
                                                buf0, (float*)d_out);
}